// ColBERTLoss_67491116089370
// MI455X (gfx1250) — compile-verified
//
#include <hip/hip_runtime.h>
#include <math.h>
#include <stdint.h>

typedef float v2f __attribute__((ext_vector_type(2)));
typedef float v8f __attribute__((ext_vector_type(8)));
typedef int   v4i_ __attribute__((vector_size(16)));

#define BSZ 64
#define SQd 64
#define SDd 512
#define Hd  128

#define CH_ROWS 64          // d-rows staged per LDS chunk (4 N-tiles of 16)
#define LDSS    132         // padded row stride (floats): 4n mod 64 bank rotation
#define CHUNKS  (SDd / CH_ROWS)
#define BUF_FLOATS (CH_ROWS * LDSS)
#define DYN_LDS_BYTES (2 * BUF_FLOATS * (int)sizeof(float))

#if defined(__HIP_DEVICE_COMPILE__) && __has_builtin(__builtin_amdgcn_global_load_async_to_lds_b128)
#define HAVE_ASYNC 1
#else
#define HAVE_ASYNC 0
#endif

#define AS1 __attribute__((address_space(1)))
#define AS3 __attribute__((address_space(3)))

__device__ __forceinline__ void copy16_g2lds(const float* gsrc, float* ldst) {
#if HAVE_ASYNC
  // Generic LDS pointer low 32 bits == LDS byte offset (shared aperture layout).
  AS1 v4i_* g = (AS1 v4i_*)(unsigned long long)(uintptr_t)gsrc;
  AS3 v4i_* l = (AS3 v4i_*)(unsigned)(uintptr_t)ldst;
  __builtin_amdgcn_global_load_async_to_lds_b128(g, l, 0, 0);
#else
  *(float4*)ldst = *(const float4*)gsrc;
#endif
}

__device__ __forceinline__ void async_wait0() {
#if HAVE_ASYNC
#if __has_builtin(__builtin_amdgcn_s_wait_asynccnt)
  __builtin_amdgcn_s_wait_asynccnt(0);
#else
  asm volatile("s_wait_asynccnt 0" ::: "memory");
#endif
#endif
}

// One block = one (a, b, side) pair. 512 threads = 16 waves.
// Wave w: M-tile m = w%4 (q rows m*16..m*16+15), N-group g = w/4.
// Per chunk of 64 d-rows there are 4 N-tiles; wave takes tile g.
__global__ __launch_bounds__(512, 1)
void colbert_maxsim_kernel(const float* __restrict__ q,
                           const float* __restrict__ dpos,
                           const float* __restrict__ dneg,
                           const int* __restrict__ mpos,
                           const int* __restrict__ mneg,
                           float* __restrict__ dist) {
  extern __shared__ float lds_dyn[];          // 2 x (64 x 132) floats, ~67.6 KB
  __shared__ float partial[4][64];
  __shared__ float red[64];

  const int b    = blockIdx.x;
  const int a    = blockIdx.y;
  const int side = blockIdx.z;
  const float* dmat = side ? dneg : dpos;
  const int*   mask = side ? mneg : mpos;

  const int tid  = threadIdx.x;
  const int w    = tid >> 5;
  const int l    = tid & 31;
  const int m    = w & 3;       // M tile group
  const int g    = w >> 2;      // N tile group
  const int h    = l >> 4;      // half of wave
  const int n    = l & 15;      // row (A) / col (B) within tile
  const int koff = 2 * h;       // K-pair selected by lane half

  // A fragments: ISA 16x4 f32 layout -> lane holds row n, K pair {4k+koff, +1}.
  const float* qrow = q + ((size_t)a * SQd + m * 16 + n) * Hd;
  v2f afrag[32];
#pragma unroll
  for (int k = 0; k < 32; ++k)
    afrag[k] = *(const v2f*)(qrow + 4 * k + koff);

  const float* dbase = dmat + (size_t)b * SDd * Hd;
  const int*   mrow  = mask + b * SDd;

  // Hoist this lane's 8 column-mask bits (column t = c*64 + g*16 + n).
  unsigned mbits = 0;
#pragma unroll
  for (int c = 0; c < CHUNKS; ++c)
    mbits |= (mrow[c * CH_ROWS + g * 16 + n] != 0 ? 1u : 0u) << c;

  float vmax[8];
#pragma unroll
  for (int r = 0; r < 8; ++r) vmax[r] = -INFINITY;

  // Cooperative async stage of a 64x128 f32 chunk (4 x b128 per thread).
  auto issue_chunk = [&](int c) {
    const float* gsrc = dbase + (size_t)c * CH_ROWS * Hd;
    float* dstb = lds_dyn + (c & 1) * BUF_FLOATS;
#pragma unroll
    for (int i = 0; i < 4; ++i) {
      int idx = tid + i * 512;           // 2048 float4s total
      int row = idx >> 5;                // 0..63
      int c4  = (idx & 31) << 2;         // 0..124
      copy16_g2lds(gsrc + row * Hd + c4, dstb + row * LDSS + c4);
    }
  };

  issue_chunk(0);
  const int tloc = g * 16 + n;           // B column n -> d row within chunk
  for (int c = 0; c < CHUNKS; ++c) {
    async_wait0();                        // own async loads for chunk c done
    __syncthreads();                      // chunk c visible to all waves
    if (c + 1 < CHUNKS) issue_chunk(c + 1);   // overlap next stage with compute

    const float* buf = lds_dyn + (c & 1) * BUF_FLOATS;
    v8f acc = {};
#pragma unroll
    for (int k = 0; k < 32; ++k) {
      // B[kk][n] = d[b][t0+n][kk]; lane reads its K pair for its column.
      v2f bfrag = *(const v2f*)(&buf[tloc * LDSS + 4 * k + koff]);
      acc = __builtin_amdgcn_wmma_f32_16x16x4_f32(false, afrag[k], false, bfrag,
                                                  (short)0, acc, false, false);
    }
    const bool valid = (mbits >> c) & 1u;
#pragma unroll
    for (int r = 0; r < 8; ++r) {
      float v = valid ? acc[r] : -INFINITY;
      vmax[r] = fmaxf(vmax[r], v);
    }
    __syncthreads();                      // done reading buf before overwrite
  }

  // Row-max across the 16 columns held in each half-wave (xor masks stay in half).
#pragma unroll
  for (int r = 0; r < 8; ++r) {
    float v = vmax[r];
    v = fmaxf(v, __shfl_xor(v, 1, 32));
    v = fmaxf(v, __shfl_xor(v, 2, 32));
    v = fmaxf(v, __shfl_xor(v, 4, 32));
    v = fmaxf(v, __shfl_xor(v, 8, 32));
    vmax[r] = v;
  }
  if (n == 0) {   // lanes 0 and 16: rows m*16 + r + 8*h
#pragma unroll
    for (int r = 0; r < 8; ++r)
      partial[g][m * 16 + r + 8 * h] = vmax[r];
  }
  __syncthreads();

  if (tid < 64) {
    float rm = fmaxf(fmaxf(partial[0][tid], partial[1][tid]),
                     fmaxf(partial[2][tid], partial[3][tid]));
    red[tid] = rm;
  }
  __syncthreads();
  for (int s = 32; s > 0; s >>= 1) {
    if (tid < s) red[tid] += red[tid + s];
    __syncthreads();
  }
  if (tid == 0)
    dist[a * (2 * BSZ) + side * BSZ + b] = red[0];
}

// logits [64][128] -> -mean_a log_softmax(row a)[a]
__global__ void colbert_loss_kernel(const float* __restrict__ dist,
                                    float* __restrict__ out) {
  __shared__ float sm[128];
  __shared__ float lacc;
  const int t = threadIdx.x;   // 128 threads
  if (t == 0) lacc = 0.f;
  __syncthreads();
  for (int a = 0; a < BSZ; ++a) {
    float v = dist[a * 128 + t];
    sm[t] = v; __syncthreads();
    for (int s = 64; s > 0; s >>= 1) {
      if (t < s) sm[t] = fmaxf(sm[t], sm[t + s]);
      __syncthreads();
    }
    float mx = sm[0]; __syncthreads();
    sm[t] = expf(v - mx); __syncthreads();
    for (int s = 64; s > 0; s >>= 1) {
      if (t < s) sm[t] += sm[t + s];
      __syncthreads();
    }
    float lse = mx + logf(sm[0]);
    __syncthreads();
    if (t == a) lacc += lse - v;   // -(v - lse)
    __syncthreads();
  }
  if (t == 0) out[0] = lacc / (float)BSZ;
}

extern "C" void kernel_launch(void* const* d_in, const int* in_sizes, int n_in,
                              void* d_out, int out_size, void* d_ws, size_t ws_size,
                              hipStream_t stream) {
  (void)in_sizes; (void)n_in; (void)out_size; (void)ws_size;
  const float* q  = (const float*)d_in[0];
  const float* dp = (const float*)d_in[1];
  const float* dn = (const float*)d_in[2];
  const int*   mp = (const int*)d_in[3];
  const int*   mn = (const int*)d_in[4];
  float* dist = (float*)d_ws;            // 64*128 floats = 32 KB scratch

  dim3 grid(BSZ, BSZ, 2);
  colbert_maxsim_kernel<<<grid, 512, DYN_LDS_BYTES, stream>>>(q, dp, dn, mp, mn, dist);
  colbert_loss_kernel<<<1, 128, 0, stream>>>(dist, (float*)d_out);
}